// myYOLOv3_73856257622116
// MI455X (gfx1250) — compile-verified
//
#include <hip/hip_runtime.h>
#include <math.h>

#ifndef __has_builtin
#define __has_builtin(x) 0
#endif

#define N_ANCH 10647
#define NP     10656   // padded to multiple of 32
#define NCLS   20

#if __has_builtin(__builtin_amdgcn_wmma_f32_16x16x4_f32)
#define HAVE_WMMA_F32X4 1
#else
#define HAVE_WMMA_F32X4 0
#endif

#if __has_builtin(__builtin_amdgcn_global_load_async_to_lds_b32)
#define HAVE_ASYNC_LDS 1
#else
#define HAVE_ASYNC_LDS 0
#endif

typedef float v2f __attribute__((ext_vector_type(2)));
typedef float v8f __attribute__((ext_vector_type(8)));

// anchors scaled by [4,2,1] per scale, per the reference _make_grids()
__constant__ float c_AW[9] = {4.f, 8.f, 12.f, 3.f, 6.f, 8.f, 3.5f, 5.f, 8.f};
__constant__ float c_AH[9] = {6.f, 12.f, 10.f, 7.f, 8.f, 6.f, 5.f, 4.5f, 8.f};

struct AnchorMap { const float* p; int HW; int s; int a; int W; float st; int sc; };

__device__ __forceinline__ AnchorMap map_anchor(int n, const float* p1,
                                                const float* p2, const float* p3) {
  AnchorMap m; int l;
  if (n < 8112)       { m.p = p1; m.HW = 2704; m.W = 52; m.st = 8.f;  m.sc = 0; l = n; }
  else if (n < 10140) { m.p = p2; m.HW = 676;  m.W = 26; m.st = 16.f; m.sc = 1; l = n - 8112; }
  else                { m.p = p3; m.HW = 169;  m.W = 13; m.st = 32.f; m.sc = 2; l = n - 10140; }
  m.s = l / 3; m.a = l - 3 * m.s;
  return m;
}

// ---------------------------------------------------------------------------
// Kernel 1: softmax denominators S[n] = sum_c exp(cls[n,c]) for 16 anchors per
// wave via chained V_WMMA_F32_16X16X4_F32 (E(16x20) x Ones(20x16): every
// column of D is the row sum). A-layout per ISA: lanes 0-15 hold K=0/1,
// lanes 16-31 hold K=2/3 of each K=4 chunk.
// ---------------------------------------------------------------------------
__global__ void yolo_cls_sums_wmma(const float* __restrict__ p1,
                                   const float* __restrict__ p2,
                                   const float* __restrict__ p3,
                                   float* __restrict__ S) {
  const int lane = threadIdx.x & 31;
  const int wave = threadIdx.x >> 5;
  const int gw   = blockIdx.x * (blockDim.x >> 5) + wave;
  const int base = gw * 16;
  const int row  = lane & 15;
  const int half = lane >> 4;          // 0: K%4 in {0,1}; 1: K%4 in {2,3}
  const int n    = base + row;

  float e[10];
  if (n < N_ANCH) {
    AnchorMap m = map_anchor(n, p1, p2, p3);
    const float* cp = m.p + (3 + m.a * NCLS) * m.HW + m.s;
#pragma unroll
    for (int st = 0; st < 5; ++st) {
      int c0 = 4 * st + 2 * half;
      e[2 * st]     = expf(cp[c0 * m.HW]);
      e[2 * st + 1] = expf(cp[(c0 + 1) * m.HW]);
    }
  } else {
#pragma unroll
    for (int q = 0; q < 10; ++q) e[q] = 0.f;   // tail rows contribute zero
  }

#if HAVE_WMMA_F32X4
  v8f acc = {0.f, 0.f, 0.f, 0.f, 0.f, 0.f, 0.f, 0.f};
  v2f ones; ones.x = 1.f; ones.y = 1.f;
#pragma unroll
  for (int st = 0; st < 5; ++st) {
    v2f a; a.x = e[2 * st]; a.y = e[2 * st + 1];
    acc = __builtin_amdgcn_wmma_f32_16x16x4_f32(
        /*neg_a=*/false, a, /*neg_b=*/false, ones,
        /*c_mod=*/(short)0, acc, /*reuse_a=*/false, /*reuse_b=*/false);
  }
  // C/D layout: VGPR r holds row r (lanes 0-15) and row 8+r (lanes 16-31);
  // all columns equal, so one predicated store per row suffices.
#pragma unroll
  for (int r = 0; r < 8; ++r) {
    if (lane == r        && (base + r)     < N_ANCH) S[base + r]     = acc[r];
    if (lane == (16 + r) && (base + 8 + r) < N_ANCH) S[base + 8 + r] = acc[r];
  }
#else
  float partial = 0.f;
#pragma unroll
  for (int q = 0; q < 10; ++q) partial += e[q];
  partial += __shfl_xor(partial, 16, 32);
  if (half == 0 && n < N_ANCH) S[n] = partial;
#endif
}

// ---------------------------------------------------------------------------
// Kernel 2: per-anchor decode (batch 0 only). Writes NMS inputs to ws and the
// bboxes/scores/cls_inds output regions.
// ---------------------------------------------------------------------------
__global__ void yolo_decode(const float* __restrict__ p1, const float* __restrict__ p2,
                            const float* __restrict__ p3, const float* __restrict__ S,
                            float* __restrict__ out,
                            float* __restrict__ ws_box, float* __restrict__ ws_score,
                            unsigned* __restrict__ ws_cls, unsigned* __restrict__ ws_valid) {
  int n = blockIdx.x * blockDim.x + threadIdx.x;
  if (n >= N_ANCH) return;
  AnchorMap m = map_anchor(n, p1, p2, p3);
  const int HW = m.HW, s = m.s, a = m.a;

  float obj = 1.f / (1.f + expf(-m.p[a * HW + s]));

  const float* cp = m.p + (3 + a * NCLS) * HW + s;
  float xm = -1e30f; int am = 0;
#pragma unroll
  for (int c = 0; c < NCLS; ++c) {
    float v = cp[c * HW];
    if (v > xm) { xm = v; am = c; }    // strict '>' keeps first max (jnp.argmax)
  }
  float score = (expf(xm) / S[n]) * obj;

  const float* xp = m.p + (63 + a * 4) * HW + s;
  float tx = xp[0], ty = xp[HW], tw = xp[2 * HW], th = xp[3 * HW];
  float gx = (float)(s % m.W), gy = (float)(s / m.W);
  float cx = 1.f / (1.f + expf(-tx)) + gx;
  float cy = 1.f / (1.f + expf(-ty)) + gy;

  float aw, ah;
  if (s == HW - 1) { aw = 0.f; ah = 0.f; }           // awh[hs*ws-1] = 0 quirk
  else             { aw = c_AW[m.sc * 3 + a]; ah = c_AH[m.sc * 3 + a]; }
  float bw = expf(tw) * aw, bh = expf(th) * ah;

  const float inv = 1.f / 416.f;
  float x1 = (cx - 0.5f * bw) * m.st * inv;
  float y1 = (cy - 0.5f * bh) * m.st * inv;
  float x2 = (cx + 0.5f * bw) * m.st * inv;
  float y2 = (cy + 0.5f * bh) * m.st * inv;

  ws_box[4 * n + 0] = x1; ws_box[4 * n + 1] = y1;
  ws_box[4 * n + 2] = x2; ws_box[4 * n + 3] = y2;
  ws_score[n] = score;
  ws_cls[n]   = (unsigned)am;
  ws_valid[n] = (score >= 0.001f) ? 1u : 0u;

  // outputs: clipped bboxes, scores, cls_inds (as float)
  out[4 * n + 0] = fminf(fmaxf(x1 * 416.f, 0.f), 415.f) * inv;
  out[4 * n + 1] = fminf(fmaxf(y1 * 416.f, 0.f), 415.f) * inv;
  out[4 * n + 2] = fminf(fmaxf(x2 * 416.f, 0.f), 415.f) * inv;
  out[4 * n + 3] = fminf(fmaxf(y2 * 416.f, 0.f), 415.f) * inv;
  out[4 * N_ANCH + n] = score;
  out[5 * N_ANCH + n] = (float)am;
}

// ---------------------------------------------------------------------------
// Kernel 3: stable descending rank (== position in jnp.argsort(-scores)) and
// scatter into sorted arrays. O(N^2) over a 42KB L1/L2-resident score array.
// ---------------------------------------------------------------------------
__global__ void yolo_rank(const float* __restrict__ score, const float* __restrict__ box,
                          const unsigned* __restrict__ cls, const unsigned* __restrict__ valid,
                          unsigned* __restrict__ rank,
                          float* __restrict__ sx1, float* __restrict__ sy1,
                          float* __restrict__ sx2, float* __restrict__ sy2,
                          unsigned* __restrict__ scv) {
  int n = blockIdx.x * blockDim.x + threadIdx.x;
  if (n >= N_ANCH) return;
  float sn = score[n];
  int r = 0;
  for (int j = 0; j < N_ANCH; ++j) {
    float sj = score[j];
    r += (sj > sn) || ((sj == sn) && (j < n));
  }
  rank[n] = (unsigned)r;
  sx1[r] = box[4 * n + 0]; sy1[r] = box[4 * n + 1];
  sx2[r] = box[4 * n + 2]; sy2[r] = box[4 * n + 3];
  scv[r] = cls[n] | (valid[n] << 8);
}

// ---------------------------------------------------------------------------
// Kernel 4: greedy NMS, fully LDS-resident (~234KB of CDNA5's 320KB WGP LDS),
// staged via async global->LDS loads when available. One workgroup, one
// barrier per greedy step; supp[j] owned by thread (j mod 1024) -> no atomics.
// Tail: scatter keep back through the rank permutation.
// ---------------------------------------------------------------------------
#if HAVE_ASYNC_LDS
// Param types per hipcc diagnostic: addrspace(1) int* (prints as "__device__
// int*") for the global source; LDS side cast to addrspace(3) int*.
#define ASYNC_G2L_B32(gsrc, ldst)                                         \
  __builtin_amdgcn_global_load_async_to_lds_b32(                          \
      (__attribute__((address_space(1))) int*)(gsrc),                     \
      (__attribute__((address_space(3))) int*)(ldst), 0, 0)
#endif

__global__ __launch_bounds__(1024)
void yolo_nms(const float* __restrict__ sx1, const float* __restrict__ sy1,
              const float* __restrict__ sx2, const float* __restrict__ sy2,
              const unsigned* __restrict__ scv, const unsigned* __restrict__ rank,
              float* __restrict__ out_keep) {
  __shared__ float lx1[NP], ly1[NP], lx2[NP], ly2[NP];
  __shared__ unsigned lscv[NP];
  __shared__ unsigned char lsupp[NP], lkeep[NP];

  const int t = threadIdx.x;

  for (int j = t; j < N_ANCH; j += 1024) {
#if HAVE_ASYNC_LDS
    ASYNC_G2L_B32(sx1 + j, lx1 + j);
    ASYNC_G2L_B32(sy1 + j, ly1 + j);
    ASYNC_G2L_B32(sx2 + j, lx2 + j);
    ASYNC_G2L_B32(sy2 + j, ly2 + j);
    ASYNC_G2L_B32(scv + j, lscv + j);
#else
    lx1[j] = sx1[j]; ly1[j] = sy1[j];
    lx2[j] = sx2[j]; ly2[j] = sy2[j];
    lscv[j] = scv[j];
#endif
    lsupp[j] = 0;
  }
#if HAVE_ASYNC_LDS
#if __has_builtin(__builtin_amdgcn_s_wait_asynccnt)
  __builtin_amdgcn_s_wait_asynccnt(0);
#else
  asm volatile("s_wait_asynccnt 0" ::: "memory");
#endif
#endif
  __syncthreads();

  for (int i = 0; i < N_ANCH; ++i) {
    unsigned cvi = lscv[i];
    bool ki = ((cvi >> 8) & 1u) && (lsupp[i] == 0);
    if (t == 0) lkeep[i] = ki ? 1 : 0;
    if (ki) {
      float bx1 = lx1[i], by1 = ly1[i], bx2 = lx2[i], by2 = ly2[i];
      float ai = (bx2 - bx1) * (by2 - by1);
      unsigned ci = cvi & 0xffu;
      for (int j = t; j < N_ANCH; j += 1024) {
        if (j > i && (lscv[j] & 0xffu) == ci) {
          float xx1 = fmaxf(bx1, lx1[j]);
          float yy1 = fmaxf(by1, ly1[j]);
          float xx2 = fminf(bx2, lx2[j]);
          float yy2 = fminf(by2, ly2[j]);
          float inter = fmaxf(1e-28f, xx2 - xx1) * fmaxf(1e-28f, yy2 - yy1);
          float aj = (lx2[j] - lx1[j]) * (ly2[j] - ly1[j]);
          float iou = inter / (ai + aj - inter);
          if (iou > 0.5f) lsupp[j] = 1;  // j mod 1024 == t: exclusive owner
        }
      }
    }
    __syncthreads();
  }

  for (int n = t; n < N_ANCH; n += 1024)
    out_keep[n] = lkeep[rank[n]] ? 1.f : 0.f;
}

// ---------------------------------------------------------------------------
extern "C" void kernel_launch(void* const* d_in, const int* in_sizes, int n_in,
                              void* d_out, int out_size, void* d_ws, size_t ws_size,
                              hipStream_t stream) {
  (void)in_sizes; (void)n_in; (void)out_size; (void)ws_size;
  const float* p1 = (const float*)d_in[0];   // (64,75,52,52) — batch 0 used
  const float* p2 = (const float*)d_in[1];   // (64,75,26,26)
  const float* p3 = (const float*)d_in[2];   // (64,75,13,13)
  float* out = (float*)d_out;                // [bboxes 4N | scores N | cls N | keep N]
  float* ws  = (float*)d_ws;

  float*    S     = ws;                       // NP
  float*    box   = ws + NP;                  // 4*NP
  float*    score = ws + 5 * NP;              // NP
  unsigned* cls   = (unsigned*)(ws + 6 * NP); // NP
  unsigned* valid = (unsigned*)(ws + 7 * NP); // NP
  unsigned* rank  = (unsigned*)(ws + 8 * NP); // NP
  float*    sx1   = ws + 9 * NP;
  float*    sy1   = ws + 10 * NP;
  float*    sx2   = ws + 11 * NP;
  float*    sy2   = ws + 12 * NP;
  unsigned* scv   = (unsigned*)(ws + 13 * NP);

  // 666 row-groups of 16 anchors -> 84 blocks x 8 waves (no early-exit: EXEC
  // stays all-ones around the WMMAs; tail lanes feed zeros)
  yolo_cls_sums_wmma<<<84, 256, 0, stream>>>(p1, p2, p3, S);
  yolo_decode<<<(N_ANCH + 255) / 256, 256, 0, stream>>>(p1, p2, p3, S, out,
                                                        box, score, cls, valid);
  yolo_rank<<<(N_ANCH + 255) / 256, 256, 0, stream>>>(score, box, cls, valid, rank,
                                                      sx1, sy1, sx2, sy2, scv);
  yolo_nms<<<1, 1024, 0, stream>>>(sx1, sy1, sx2, sy2, scv, rank, out + 6 * N_ANCH);
}